// CartesianEquivariantBasisBlock_88201448390766
// MI455X (gfx1250) — compile-verified
//
#include <hip/hip_runtime.h>
#include <hip/hip_bf16.h>

typedef __attribute__((ext_vector_type(2))) float v2f;
typedef __attribute__((ext_vector_type(8))) float v8f;

// One wave32 computes everything. The rank-2 (3x3) and rank-3 (3x3x3)
// outer products are fused into a single rank-1 V_WMMA_F32_16X16X4_F32:
//   A (16x4): K=0 column = u (rows 0..2), rest 0
//   B (4x16): K=0 row    = [u0,u1,u2, q0..q8], q_t = u_{t/3}*u_{t%3}, rest 0
//   D[m][n]  = u_m * B0[n]  ->  n<3: u_m*u_n ; n=3..11: u_m*u_j*u_k
__global__ __launch_bounds__(32) void equiv_basis_kernel(
    const float* __restrict__ u,
    const float* __restrict__ w0, const float* __restrict__ w1,
    const float* __restrict__ w2, const float* __restrict__ w3,
    float* __restrict__ out)
{
    const int lane = threadIdx.x;

    const float u0 = u[0], u1 = u[1], u2 = u[2];
    const float dot  = u0 * u0 + u1 * u1 + u2 * u2;
    const float dot2 = dot * dot;
    const float dot3 = dot2 * dot;

    // Uniform weight-segment sums (broadcast loads, redundant per-lane is fine).
    float s0a = w0[0], s0b = 0.f, s0c = 0.f;
    for (int i = 1; i < 4;  ++i) s0b += w0[i];
    for (int i = 4; i < 19; ++i) s0c += w0[i];
    float s1a = w1[0], s1b = 0.f, s1c = 0.f;
    for (int i = 1; i < 4;  ++i) s1b += w1[i];
    for (int i = 4; i < 19; ++i) s1c += w1[i];
    float s2a = w2[0], s2b = 0.f, s2c = 0.f;
    for (int i = 1; i < 7;  ++i) s2b += w2[i];
    for (int i = 7; i < 52; ++i) s2c += w2[i];
    float s3a = w3[0], s3b = 0.f;
    for (int i = 1; i < 11; ++i) s3b += w3[i];

    const float c0 = s0a * dot + s0b * dot2 + s0c * dot3;
    const float c1 = s1a + s1b * dot + s1c * dot2;
    const float c2 = s2a + s2b * dot + s2c * dot2;
    const float c3 = s3a + s3b * dot;

    const float uv[3] = { u0, u1, u2 };

    // Build WMMA operands. A-matrix 16x4 f32: lane m (<16) VGPR0 = K=0 elem of
    // row m; lanes 16..31 / second VGPR hold K=2 / K=1,3 -> all zero here.
    v2f a; a.x = 0.f; a.y = 0.f;
    if (lane < 3) a.x = uv[lane];

    // B-matrix 4x16 f32: row K striped across lanes within a VGPR; we only
    // populate row K=0 (lanes 0..15 of the first VGPR), everything else zero,
    // so any K>0 slot placement contributes nothing.
    v2f b; b.x = 0.f; b.y = 0.f;
    if (lane < 3) {
        b.x = uv[lane];
    } else if (lane < 12) {
        const int t = lane - 3;
        b.x = uv[t / 3] * uv[t % 3];
    }

    v8f c = {};
    // (neg_a, A, neg_b, B, c_mod, C, reuse_a, reuse_b)
    v8f d = __builtin_amdgcn_wmma_f32_16x16x4_f32(
        false, a, false, b, (short)0, c, false, false);

    // D layout (16x16 f32 C/D): lane l<16 -> N=l, VGPR v -> M=v.
    // out layout: [0]=scalar, [1..3]=c1*u, [4..12]=c2*uuT (row major),
    //             [13..39]=c3*u(x)u(x)u (row major).
    if (lane == 0) out[0] = c0;
    if (lane < 3) {
        out[1 + lane]     = c1 * uv[lane];
        out[4 + 0 + lane] = c2 * d[0];   // m=0 row of uuT
        out[4 + 3 + lane] = c2 * d[1];   // m=1
        out[4 + 6 + lane] = c2 * d[2];   // m=2
    } else if (lane < 12) {
        const int t = lane - 3;          // t = j*3 + k
        out[13 + 0  + t] = c3 * d[0];    // m=0 slab
        out[13 + 9  + t] = c3 * d[1];    // m=1
        out[13 + 18 + t] = c3 * d[2];    // m=2
    }
}

extern "C" void kernel_launch(void* const* d_in, const int* in_sizes, int n_in,
                              void* d_out, int out_size, void* d_ws, size_t ws_size,
                              hipStream_t stream) {
    (void)in_sizes; (void)n_in; (void)out_size; (void)d_ws; (void)ws_size;
    const float* u  = (const float*)d_in[0];
    const float* w0 = (const float*)d_in[1];
    const float* w1 = (const float*)d_in[2];
    const float* w2 = (const float*)d_in[3];
    const float* w3 = (const float*)d_in[4];
    float* out = (float*)d_out;
    equiv_basis_kernel<<<1, 32, 0, stream>>>(u, w0, w1, w2, w3, out);
}